// ObjectCaps_6236292513948
// MI455X (gfx1250) — compile-verified
//
#include <hip/hip_runtime.h>
#include <hip/hip_bf16.h>
#include <math.h>

typedef __attribute__((ext_vector_type(2))) float v2f;
typedef __attribute__((ext_vector_type(8))) float v8f;

#define NROUTES 2048
#define IC 32
#define CAPS 32
#define OC 32
#define BATCH 64
#define MTOT 1024            // CAPS*OC
#define SLAB 65536           // MTOT*BATCH  (one route's u_hat slab, [m][b])

// ---------------------------------------------------------------------------
// Kernel 1: u_hat[r][m][b] = sum_i W[r][m][i] * x[b][r][i]   via fp32 WMMA
// One block per route. 8 waves; wave w owns M-tiles 8w..8w+7 (16 rows each),
// all 4 N-tiles (b 0..63), K = 32 in 8 steps of 4.
// ---------------------------------------------------------------------------
__global__ __launch_bounds__(256)
void uhat_gemm(const float* __restrict__ x, const float* __restrict__ W,
               float* __restrict__ u) {
  const int r    = blockIdx.x;
  const int t    = threadIdx.x;
  const int wave = t >> 5;
  const int lane = t & 31;
  const int half = lane >> 4;     // 0: lanes 0-15, 1: lanes 16-31
  const int l15  = lane & 15;

  // xT[i][b] with padded stride 65 (bank-conflict free)
  __shared__ float xT[IC * 65];
  for (int k = t; k < IC * BATCH; k += 256) {
    int b = k >> 5;               // coalesced global read over i
    int i = k & 31;
    xT[i * 65 + b] = x[(size_t)b * (NROUTES * IC) + (size_t)r * IC + i];
  }
  __syncthreads();

  // Preload all B fragments (shared by every M-tile of this wave).
  // B is 4x16 (KxN): lanes 0-15 hold rows K0/K1, lanes 16-31 rows K2/K3.
  v2f bfrag[8][4];
#pragma unroll
  for (int kk = 0; kk < 8; ++kk)
#pragma unroll
    for (int nt = 0; nt < 4; ++nt) {
      int i0   = 4 * kk + 2 * half;
      int bcol = 16 * nt + l15;
      bfrag[kk][nt][0] = xT[i0 * 65 + bcol];
      bfrag[kk][nt][1] = xT[(i0 + 1) * 65 + bcol];
    }

  const float* Wr = W + (size_t)r * (MTOT * IC);
  float*       ur = u + (size_t)r * SLAB;

  for (int mi = 0; mi < 8; ++mi) {
    int mt   = wave * 8 + mi;
    int mrow = mt * 16 + l15;     // A-row held by this lane
    const float* wrow = Wr + (size_t)mrow * IC + 2 * half;  // K base per half
    v8f acc[4] = {};
#pragma unroll
    for (int kk = 0; kk < 8; ++kk) {
      v2f a;
      a[0] = wrow[4 * kk + 0];    // A 16x4: lanes<16 K=0,1 ; lanes>=16 K=2,3
      a[1] = wrow[4 * kk + 1];
#pragma unroll
      for (int nt = 0; nt < 4; ++nt) {
        acc[nt] = __builtin_amdgcn_wmma_f32_16x16x4_f32(
            false, a, false, bfrag[kk][nt], (short)0, acc[nt], false, false);
      }
    }
    // D 16x16: VGPR j -> M = 16*mt + j + 8*half, N = 16*nt + l15
#pragma unroll
    for (int nt = 0; nt < 4; ++nt) {
      int ncol = 16 * nt + l15;
#pragma unroll
      for (int j = 0; j < 8; ++j) {
        int Mrow = mt * 16 + j + 8 * half;
        ur[(size_t)Mrow * BATCH + ncol] = acc[nt][j];
      }
    }
  }
}

// ---------------------------------------------------------------------------
// init: c_ij = 1/R (softmax of zero logits), b_ij = 0
// ---------------------------------------------------------------------------
__global__ void init_k(float* __restrict__ cij, float* __restrict__ bij) {
  int tid = blockIdx.x * 256 + threadIdx.x;   // 65536 = R*CAPS
  cij[tid] = 1.0f / (float)NROUTES;
  bij[tid] = 0.0f;
}

// ---------------------------------------------------------------------------
// s-pass: s_part[chunk][m*64+b] = sum_{r in chunk} c_ij[r,c] * u[r][m][b]
// thread <-> (m,b); fully coalesced streaming of u.
// ---------------------------------------------------------------------------
__global__ __launch_bounds__(256)
void s_pass(const float* __restrict__ u, const float* __restrict__ cij,
            float* __restrict__ s_part) {
  int tid   = blockIdx.x * 256 + threadIdx.x;  // [0, 65536)
  int chunk = blockIdx.y;                      // [0, 16)
  int c     = tid >> 11;                       // m = tid>>6, c = m>>5
  float acc = 0.0f;
  int r0 = chunk * (NROUTES / 16);
  for (int rr = 0; rr < NROUTES / 16; ++rr) {
    int r = r0 + rr;
    acc += cij[r * CAPS + c] * u[(size_t)r * SLAB + tid];
  }
  s_part[(size_t)chunk * SLAB + tid] = acc;
}

// ---------------------------------------------------------------------------
// squash: v[m*64+b] = sq*s / ((1+sq)*sqrt(sq)), sq = s*s  (elementwise, faithful)
// ---------------------------------------------------------------------------
__global__ void squash_k(const float* __restrict__ s_part, float* __restrict__ v) {
  int tid = blockIdx.x * 256 + threadIdx.x;
  float s = 0.0f;
  for (int k = 0; k < 16; ++k) s += s_part[(size_t)k * SLAB + tid];
  float sq = s * s;
  v[tid] = sq * s / ((1.0f + sq) * sqrtf(sq));
}

// ---------------------------------------------------------------------------
// a-pass: b_ij[r,c] += (1/64) * sum_b sum_o u[r][c,o][b] * v[c,o][b]
// one block per route; groups of 64 threads reduce over b for each cap.
// ---------------------------------------------------------------------------
__global__ __launch_bounds__(256)
void a_pass(const float* __restrict__ u, const float* __restrict__ v,
            float* __restrict__ bij) {
  int r = blockIdx.x;
  int t = threadIdx.x;
  int b   = t & 63;
  int grp = t >> 6;                // 4 cap-groups
  __shared__ float red[256];
  const float* ur = u + (size_t)r * SLAB;
  for (int c = grp; c < CAPS; c += 4) {
    const float* uc = ur + c * (OC * BATCH);
    const float* vc = v  + c * (OC * BATCH);
    float acc = 0.0f;
    for (int o = 0; o < OC; ++o)
      acc += uc[o * BATCH + b] * vc[o * BATCH + b];
    red[t] = acc;
    __syncthreads();
    for (int s = 32; s >= 1; s >>= 1) {
      if ((t & 63) < s) red[t] += red[t + s];
      __syncthreads();
    }
    if (b == 0) bij[r * CAPS + c] += red[t] * (1.0f / 64.0f);
    __syncthreads();
  }
}

// ---------------------------------------------------------------------------
// softmax over routes (axis 0) per capsule column
// ---------------------------------------------------------------------------
__global__ __launch_bounds__(256)
void softmax_k(const float* __restrict__ bij, float* __restrict__ cij) {
  int c = blockIdx.x;
  int t = threadIdx.x;
  __shared__ float red[256];
  float m = -3.402823466e+38f;
  for (int r = t; r < NROUTES; r += 256) m = fmaxf(m, bij[r * CAPS + c]);
  red[t] = m; __syncthreads();
  for (int s = 128; s >= 1; s >>= 1) {
    if (t < s) red[t] = fmaxf(red[t], red[t + s]);
    __syncthreads();
  }
  m = red[0]; __syncthreads();
  float sum = 0.0f;
  for (int r = t; r < NROUTES; r += 256) sum += expf(bij[r * CAPS + c] - m);
  red[t] = sum; __syncthreads();
  for (int s = 128; s >= 1; s >>= 1) {
    if (t < s) red[t] += red[t + s];
    __syncthreads();
  }
  float inv = 1.0f / red[0];
  for (int r = t; r < NROUTES; r += 256)
    cij[r * CAPS + c] = expf(bij[r * CAPS + c] - m) * inv;
}

// ---------------------------------------------------------------------------
// output: d_out[b, c, o, 1] = v[(c*32+o)*64 + b]
// ---------------------------------------------------------------------------
__global__ void out_k(const float* __restrict__ v, float* __restrict__ out) {
  int tid = blockIdx.x * 256 + threadIdx.x;   // b*1024 + m
  int b = tid >> 10;
  int m = tid & 1023;
  out[tid] = v[m * BATCH + b];
}

extern "C" void kernel_launch(void* const* d_in, const int* in_sizes, int n_in,
                              void* d_out, int out_size, void* d_ws, size_t ws_size,
                              hipStream_t stream) {
  (void)in_sizes; (void)n_in; (void)out_size; (void)ws_size;
  const float* x = (const float*)d_in[0];   // [B, R, IC]
  const float* W = (const float*)d_in[1];   // [R, CAPS, OC, IC]
  float* out = (float*)d_out;               // [B, CAPS, OC, 1]

  float* ws   = (float*)d_ws;
  float* u    = ws;                                   // 134,217,728 floats
  float* cij  = u   + (size_t)NROUTES * SLAB;         // 65536
  float* bij  = cij + NROUTES * CAPS;                 // 65536
  float* sp   = bij + NROUTES * CAPS;                 // 16*65536
  float* v    = sp  + 16 * SLAB;                      // 65536

  init_k<<<256, 256, 0, stream>>>(cij, bij);
  uhat_gemm<<<NROUTES, 256, 0, stream>>>(x, W, u);

  // iteration 0 (c_ij uniform), 1, 2
  for (int it = 0; it < 3; ++it) {
    if (it > 0) softmax_k<<<CAPS, 256, 0, stream>>>(bij, cij);
    s_pass<<<dim3(256, 16), 256, 0, stream>>>(u, cij, sp);
    squash_k<<<256, 256, 0, stream>>>(sp, v);
    if (it < 2) a_pass<<<NROUTES, 256, 0, stream>>>(u, v, bij);
  }
  out_k<<<256, 256, 0, stream>>>(v, out);
}